// DeltaVolumeDecoder_60550448939626
// MI455X (gfx1250) — compile-verified
//
#include <hip/hip_runtime.h>
#include <hip/hip_bf16.h>
#include <math.h>

typedef float v2f __attribute__((ext_vector_type(2)));
typedef float v8f __attribute__((ext_vector_type(8)));

#define B_   16
#define LAT_ 64
#define TV_  200000
#define V_   128
#define V3_  (V_ * V_ * V_)          // 2097152
#define NCOL (4 * TV_)               // 800000 columns of Wout

__device__ __forceinline__ float bf_round(float v) {
    return __bfloat162float(__float2bfloat16(v));
}
__device__ __forceinline__ float bf2f(__hip_bfloat16 v) { return __bfloat162float(v); }

// ---------------------------------------------------------------------------
// Kernel 1: zero the 16 x 128^3 output grids (134 MB) with wide stores.
// ---------------------------------------------------------------------------
__global__ __launch_bounds__(256) void zero_kernel(float4* __restrict__ out, int n4) {
    int i = blockIdx.x * blockDim.x + threadIdx.x;
    int stride = gridDim.x * blockDim.x;
    for (; i < n4; i += stride) out[i] = make_float4(0.f, 0.f, 0.f, 0.f);
}

// ---------------------------------------------------------------------------
// Kernel 2: tiny bf16 MLP head. One block, 128 threads = one (b,k) element
// each. Emulates JAX bf16 matmul chain (f32 accumulate, bf16 round per op).
// Result h (16x8) written to workspace as f32.
// ---------------------------------------------------------------------------
__global__ __launch_bounds__(128) void mlp_head_kernel(
    const float* __restrict__ x,
    const __hip_bfloat16* __restrict__ W0, const __hip_bfloat16* __restrict__ b0,
    const __hip_bfloat16* __restrict__ W1, const __hip_bfloat16* __restrict__ b1,
    const __hip_bfloat16* __restrict__ W2, const __hip_bfloat16* __restrict__ b2,
    const __hip_bfloat16* __restrict__ W3, const __hip_bfloat16* __restrict__ b3,
    const __hip_bfloat16* __restrict__ W4, const __hip_bfloat16* __restrict__ b4,
    float* __restrict__ h_out)
{
    __shared__ float h[B_][8];
    const int t = threadIdx.x;
    const int b = t >> 3;
    const int k = t & 7;

    // layer 0: sin(30 * (bf16(x) @ W0 + b0))
    float acc = 0.f;
    for (int i = 0; i < LAT_; ++i) {
        float xb = bf_round(x[b * LAT_ + i]);
        acc += xb * bf2f(W0[i * 8 + k]);
    }
    float pre = bf_round(acc);
    pre = bf_round(pre + bf2f(b0[k]));
    pre = bf_round(30.f * pre);
    h[b][k] = bf_round(sinf(pre));
    __syncthreads();

    const __hip_bfloat16* Ws[4] = {W1, W2, W3, W4};
    const __hip_bfloat16* bs[4] = {b1, b2, b3, b4};
    for (int l = 0; l < 4; ++l) {
        float a = 0.f;
        for (int j = 0; j < 8; ++j) a += h[b][j] * bf2f(Ws[l][j * 8 + k]);
        a = bf_round(a);
        a = bf_round(a + bf2f(bs[l][k]));
        float sn = bf_round(sinf(a));
        float hn = bf_round(h[b][k] + sn);
        __syncthreads();
        h[b][k] = hn;
        __syncthreads();
    }
    h_out[b * 8 + k] = h[b][k];
}

// ---------------------------------------------------------------------------
// Kernel 3: per wave -> one 16x16 tile of out = h @ Wout + bout via two
// V_WMMA_F32_16X16X4_F32 (K = 8 = 2*4), then trilinear scatter of the
// 4 tvs x 16 batches in that tile (2 (b,tv) pairs per lane, 8 atomics each).
// ---------------------------------------------------------------------------
__global__ __launch_bounds__(256) void decode_scatter_kernel(
    const float* __restrict__ hmat,    // [16][8]
    const float* __restrict__ Wout,    // [8][NCOL]
    const float* __restrict__ bout,    // [NCOL]
    const float* __restrict__ refv,    // [TV_]
    const int*   __restrict__ inds,    // [TV_][3]
    float* __restrict__ out)           // [16][V3_]
{
    __shared__ float tile[8][16][16];  // per-wave 16x16 out tile

    const int lane = threadIdx.x & 31;
    const int wib  = threadIdx.x >> 5;                 // wave in block (0..7)
    const int wave = blockIdx.x * 8 + wib;             // global wave id
    const int j0   = wave * 16;                        // column base in Wout
    const int hi   = lane >> 4;                        // half-wave: 0 or 1
    const int l16  = lane & 15;

    // A fragments (16x4 f32): VGPR0 = K 0/2, VGPR1 = K 1/3 per half-wave
    v2f a0, a1;
    a0.x = hmat[l16 * 8 + 0 + hi * 2];
    a0.y = hmat[l16 * 8 + 1 + hi * 2];
    a1.x = hmat[l16 * 8 + 4 + hi * 2];
    a1.y = hmat[l16 * 8 + 5 + hi * 2];

    // B fragments (4x16 f32): mirrored layout over N; coalesced 64B rows
    v2f bb0, bb1;
    bb0.x = Wout[(size_t)(0 + hi * 2) * NCOL + j0 + l16];
    bb0.y = Wout[(size_t)(1 + hi * 2) * NCOL + j0 + l16];
    bb1.x = Wout[(size_t)(4 + hi * 2) * NCOL + j0 + l16];
    bb1.y = Wout[(size_t)(5 + hi * 2) * NCOL + j0 + l16];

    // C init with bias (per-column, replicated over the 8 row-registers)
    v8f c;
    float bias = bout[j0 + l16];
    for (int v = 0; v < 8; ++v) c[v] = bias;

    c = __builtin_amdgcn_wmma_f32_16x16x4_f32(false, a0, false, bb0, (short)0, c, false, false);
    c = __builtin_amdgcn_wmma_f32_16x16x4_f32(false, a1, false, bb1, (short)0, c, false, false);

    // spill tile: component v holds row M = v + hi*8, col N = l16
    for (int v = 0; v < 8; ++v) tile[wib][v + hi * 8][l16] = c[v];
    __syncthreads();

    // scatter: 64 (b,tv) pairs per wave, 2 per lane
    for (int pp = 0; pp < 2; ++pp) {
        const int p   = lane * 2 + pp;
        const int b   = p >> 2;
        const int tvl = p & 3;
        const float4 o4 = *(const float4*)&tile[wib][b][tvl * 4];

        const int tv = wave * 4 + tvl;
        const int i0 = inds[tv * 3 + 0];
        const int i1 = inds[tv * 3 + 1];
        const int i2 = inds[tv * 3 + 2];

        // coords0 = (inds[:,2], inds[:,1], inds[:,0]); coords = coords0 + 64*delta
        const float c0 = (float)i2 + 64.f * o4.x;
        const float c1 = (float)i1 + 64.f * o4.y;
        const float c2 = (float)i0 + 64.f * o4.z;
        const float val = refv[tv] + o4.w;

        const float f0 = floorf(c0), f1 = floorf(c1), f2 = floorf(c2);
        const int   p0 = (int)f0,    p1 = (int)f1,    p2 = (int)f2;
        const float g0 = c0 - f0,    g1 = c1 - f1,    g2 = c2 - f2;

        float* __restrict__ gout = out + (size_t)b * V3_;
        #pragma unroll
        for (int o = 0; o < 8; ++o) {
            const int o0 = o & 1, o1 = (o >> 1) & 1, o2 = (o >> 2) & 1;
            const float w = (o0 ? g0 : 1.f - g0) *
                            (o1 ? g1 : 1.f - g1) *
                            (o2 ? g2 : 1.f - g2);
            const int idx0 = p0 + o0;
            const int idx1 = p1 + o1;
            const int idx2 = p2 + o2;
            const int flat = (idx2 * V_ + idx1) * V_ + idx0;  // JAX drop: flat OOB only
            if ((unsigned)flat < (unsigned)V3_) {
                atomicAdd(gout + flat, val * w);
            }
        }
    }
}

// ---------------------------------------------------------------------------
extern "C" void kernel_launch(void* const* d_in, const int* in_sizes, int n_in,
                              void* d_out, int out_size, void* d_ws, size_t ws_size,
                              hipStream_t stream) {
    const float*          x    = (const float*)d_in[0];
    const __hip_bfloat16* W0   = (const __hip_bfloat16*)d_in[1];
    const __hip_bfloat16* b0   = (const __hip_bfloat16*)d_in[2];
    const __hip_bfloat16* W1   = (const __hip_bfloat16*)d_in[3];
    const __hip_bfloat16* b1   = (const __hip_bfloat16*)d_in[4];
    const __hip_bfloat16* W2   = (const __hip_bfloat16*)d_in[5];
    const __hip_bfloat16* b2   = (const __hip_bfloat16*)d_in[6];
    const __hip_bfloat16* W3   = (const __hip_bfloat16*)d_in[7];
    const __hip_bfloat16* b3   = (const __hip_bfloat16*)d_in[8];
    const __hip_bfloat16* W4   = (const __hip_bfloat16*)d_in[9];
    const __hip_bfloat16* b4   = (const __hip_bfloat16*)d_in[10];
    const float*          Wout = (const float*)d_in[11];
    const float*          bout = (const float*)d_in[12];
    const float*          refv = (const float*)d_in[13];
    const int*            inds = (const int*)d_in[14];

    float* out = (float*)d_out;
    float* h   = (float*)d_ws;   // 16*8 f32 scratch

    // 1) zero the output grids (out_size = 16 * 128^3, divisible by 4)
    const int n4 = out_size / 4;
    int zblocks = (n4 + 255) / 256;
    if (zblocks > 16384) zblocks = 16384;   // grid-stride
    zero_kernel<<<zblocks, 256, 0, stream>>>((float4*)out, n4);

    // 2) MLP head -> h
    mlp_head_kernel<<<1, 128, 0, stream>>>(x, W0, b0, W1, b1, W2, b2, W3, b3, W4, b4, h);

    // 3) WMMA projection + trilinear scatter: 50000 waves, 8 per block
    const int nwaves = TV_ / 4;             // 50000
    decode_scatter_kernel<<<nwaves / 8, 256, 0, stream>>>(h, Wout, bout, refv, inds, out);
}